// NKATHamiltonian_18064632447059
// MI455X (gfx1250) — compile-verified
//
#include <hip/hip_runtime.h>
#include <cmath>

// ---------------------------------------------------------------------------
// NKAT Hamiltonian assembly: out = diag(h_local) + scale * (scatter + mirror)
// Pure memory-bound op: 256 MB output, ~19 MB input streams, ~1.6M FLOPs.
// Strategy:
//   1) nkat_zero_fill : grid-stride b128 (float4) zero stores, default RT
//      temporal policy so the matrix stays resident in the 192 MB L2.
//   2) nkat_diag      : 8192 diagonal stores (strict-lower scatter never
//      touches the diagonal, so this can run before the scatter).
//   3) nkat_scatter   : one thread per pair; NT loads for the read-once
//      index/value streams (protect L2 matrix residency), two 4B stores
//      (i,j) and (j,i) that mostly hit lines still resident from step 1.
// Same-stream launches give the required ordering.
// ---------------------------------------------------------------------------

__global__ __launch_bounds__(256) void nkat_zero_fill(float4* __restrict__ out4,
                                                      size_t n4,
                                                      float* __restrict__ out,
                                                      size_t n) {
    size_t tid    = (size_t)blockIdx.x * blockDim.x + threadIdx.x;
    size_t stride = (size_t)gridDim.x * blockDim.x;
    const float4 z = make_float4(0.0f, 0.0f, 0.0f, 0.0f);
    // Bulk: 16-byte stores, coalesced across the wave32 -> global_store_b128.
    for (size_t k = tid; k < n4; k += stride) {
        out4[k] = z;
    }
    // Tail (n not a multiple of 4; for D=8192 this loop body never runs).
    for (size_t k = n4 * 4 + tid; k < n; k += stride) {
        out[k] = 0.0f;
    }
}

__global__ __launch_bounds__(256) void nkat_diag(float* __restrict__ out,
                                                 const float* __restrict__ h,
                                                 int d) {
    int t = blockIdx.x * blockDim.x + threadIdx.x;
    if (t < d) {
        out[(size_t)t * (size_t)(d + 1)] = h[t];
    }
}

__global__ __launch_bounds__(256) void nkat_scatter(float* __restrict__ out,
                                                    const int* __restrict__ idx,
                                                    const float* __restrict__ v,
                                                    int m, int d, float scale) {
    int k = blockIdx.x * blockDim.x + threadIdx.x;
    if (k >= m) return;
    // Read-once streams: non-temporal loads so they don't evict the matrix
    // lines we deliberately parked in L2 during the zero fill.
    int   i   = __builtin_nontemporal_load(idx + k);        // row (i > j)
    int   j   = __builtin_nontemporal_load(idx + m + k);    // col
    float val = __builtin_nontemporal_load(v + k) * scale;
    // Unique indices -> plain stores (set semantics), symmetric pair.
    out[(size_t)i * (size_t)d + (size_t)j] = val;
    out[(size_t)j * (size_t)d + (size_t)i] = val;
}

extern "C" void kernel_launch(void* const* d_in, const int* in_sizes, int n_in,
                              void* d_out, int out_size, void* d_ws, size_t ws_size,
                              hipStream_t stream) {
    const float* h_local = (const float*)d_in[0];   // [D]   f32
    const float* V       = (const float*)d_in[1];   // [M]   f32
    const int*   idx     = (const int*)d_in[2];     // [2,M] i32 (row-major: i's then j's)
    // d_in[3] = dimension scalar; D is recovered from in_sizes instead.

    const int d = in_sizes[0];
    const int m = in_sizes[1];
    float* out = (float*)d_out;

    const size_t n  = (size_t)d * (size_t)d;
    const size_t n4 = n / 4;
    const float scale = (float)(1.0 - 0.2 / std::sqrt(std::log((double)d)));

    const int T = 256;                     // 8 wave32 waves per block
    const size_t per_thread = 8;           // 8 x b128 stores per thread
    size_t want_blocks = (n4 + (size_t)T * per_thread - 1) / ((size_t)T * per_thread);
    int fill_blocks = (int)(want_blocks < 1 ? 1 : want_blocks);

    nkat_zero_fill<<<fill_blocks, T, 0, stream>>>((float4*)out, n4, out, n);
    nkat_diag<<<(d + T - 1) / T, T, 0, stream>>>(out, h_local, d);
    nkat_scatter<<<(m + T - 1) / T, T, 0, stream>>>(out, idx, V, m, d, scale);
}